// QuantLinear_34136400068904
// MI455X (gfx1250) — compile-verified
//
#include <hip/hip_runtime.h>

typedef float v2f __attribute__((ext_vector_type(2)));
typedef float v8f __attribute__((ext_vector_type(8)));

#define TPW 16         // 16-row tiles processed per wave
#define WAVES_PER_BLOCK 8

static __device__ __forceinline__ float sgnf(float v) {
    return (v > 0.f) ? 1.f : ((v < 0.f) ? -1.f : 0.f);
}
static __device__ __forceinline__ float clampf(float v, float lo, float hi) {
    return __builtin_fminf(__builtin_fmaxf(v, lo), hi);
}

__global__ __launch_bounds__(256) void quantlinear_wmma_kernel(
    const float* __restrict__ x,       // (B,16)
    const float* __restrict__ w,       // (32,16)
    const float* __restrict__ p_in_max,
    const float* __restrict__ p_in_min,
    float* __restrict__ out,           // (B,32)
    int nTiles)
{
    const int lane = threadIdx.x & 31;
    const int wid  = (int)((blockIdx.x * blockDim.x + threadIdx.x) >> 5);

    const float in_max = p_in_max[0];
    const float in_min = p_in_min[0];
    const float rng = in_max - in_min;
    const float qs  = 15.0f / rng;         // input quantization scale
    const float qo  = -in_min * qs;        // input quantization offset

    const int nsel = lane & 15;            // M (for A) or N (for B/D) selector
    const int hi   = lane >> 4;            // half-wave id

    // ---- per-wave weight fragments: B matrices (4x16) for WMMA 16x16x4 ----
    // Weights pre-scaled by 1/15 so the WMMA directly yields y/15 (exact for
    // rint: |y/15 - half-integer| >= 1/30 >> f32 accumulation error).
    v2f bp[4][2], bm[4][2];
    float bias[2];
    #pragma unroll
    for (int h = 0; h < 2; ++h) {
        const int o = h * 16 + nsel;       // output column this lane owns
        float ssum = 0.f;
        #pragma unroll
        for (int i = 0; i < 16; ++i) ssum += sgnf(w[o * 16 + i]);
        bias[h] = in_min * ssum;           // (ones*in_min) @ sign(W)^T
        #pragma unroll
        for (int g = 0; g < 4; ++g) {
            const int kb = g * 4 + hi * 2; // K base for this half-wave
            const float s0 = sgnf(w[o * 16 + kb]);
            const float s1 = sgnf(w[o * 16 + kb + 1]);
            v2f p; p.x = (s0 > 0.f) ? (1.0f / 15.0f) : 0.f;
                   p.y = (s1 > 0.f) ? (1.0f / 15.0f) : 0.f;
            v2f m; m.x = (s0 < 0.f) ? (1.0f / 15.0f) : 0.f;
                   m.y = (s1 < 0.f) ? (1.0f / 15.0f) : 0.f;
            bp[g][h] = p;
            bm[g][h] = m;
        }
    }

    for (int it = 0; it < TPW; ++it) {
        const int t = wid * TPW + it;      // wave-uniform -> EXEC stays full
        if (t >= nTiles) return;
        const int row0 = t * 16;

        // ---- A fragments: quantized x tile, one 16x4 fragment per K-group --
        v2f a[4];
        #pragma unroll
        for (int g = 0; g < 4; ++g) {
            const int kb = g * 4 + hi * 2;
            const v2f v = __builtin_nontemporal_load(
                (const v2f*)(x + (size_t)(row0 + nsel) * 16 + kb));
            v2f q;
            q.x = clampf(__builtin_rintf(v.x * qs + qo), 0.f, 15.f);
            q.y = clampf(__builtin_rintf(v.y * qs + qo), 0.f, 15.f);
            a[g] = q;
        }

        // ---- 4 independent chains: {plus,minus} x {half0,half1}. Requant
        // clamp is provably a no-op (y/15 in [0,4]); each group contributes
        // rint() of the pre-scaled WMMA output. Group 0 is peeled so the
        // accumulators are *initialized* from it (no wasted 0+x adds).
        v8f zero = {0.f,0.f,0.f,0.f,0.f,0.f,0.f,0.f};
        v8f accP0, accM0, accP1, accM1;
        {
            v8f yp0 = __builtin_amdgcn_wmma_f32_16x16x4_f32(
                false, a[0], false, bp[0][0], (short)0, zero, false, false);
            v8f ym0 = __builtin_amdgcn_wmma_f32_16x16x4_f32(
                false, a[0], false, bm[0][0], (short)0, zero, false, false);
            v8f yp1 = __builtin_amdgcn_wmma_f32_16x16x4_f32(
                false, a[0], false, bp[0][1], (short)0, zero, false, false);
            v8f ym1 = __builtin_amdgcn_wmma_f32_16x16x4_f32(
                false, a[0], false, bm[0][1], (short)0, zero, false, false);
            #pragma unroll
            for (int j = 0; j < 8; ++j) {
                accP0[j] = __builtin_rintf(yp0[j]);
                accM0[j] = __builtin_rintf(ym0[j]);
                accP1[j] = __builtin_rintf(yp1[j]);
                accM1[j] = __builtin_rintf(ym1[j]);
            }
        }
        #pragma unroll
        for (int g = 1; g < 4; ++g) {
            v8f yp0 = __builtin_amdgcn_wmma_f32_16x16x4_f32(
                false, a[g], false, bp[g][0], (short)0, zero, false, false);
            v8f ym0 = __builtin_amdgcn_wmma_f32_16x16x4_f32(
                false, a[g], false, bm[g][0], (short)0, zero, false, false);
            v8f yp1 = __builtin_amdgcn_wmma_f32_16x16x4_f32(
                false, a[g], false, bp[g][1], (short)0, zero, false, false);
            v8f ym1 = __builtin_amdgcn_wmma_f32_16x16x4_f32(
                false, a[g], false, bm[g][1], (short)0, zero, false, false);
            #pragma unroll
            for (int j = 0; j < 8; ++j) {
                accP0[j] += __builtin_rintf(yp0[j]);
                accM0[j] += __builtin_rintf(ym0[j]);
                accP1[j] += __builtin_rintf(yp1[j]);
                accM1[j] += __builtin_rintf(ym1[j]);
            }
        }

        // D layout: elem j -> row = row0 + j + 8*hi, col = h*16 + nsel
        #pragma unroll
        for (int j = 0; j < 8; ++j) {
            const int r = row0 + j + hi * 8;
            __builtin_nontemporal_store((accP0[j] - accM0[j]) * rng + bias[0],
                                        out + (size_t)r * 32 + nsel);
            __builtin_nontemporal_store((accP1[j] - accM1[j]) * rng + bias[1],
                                        out + (size_t)r * 32 + 16 + nsel);
        }
    }
}

extern "C" void kernel_launch(void* const* d_in, const int* in_sizes, int n_in,
                              void* d_out, int out_size, void* d_ws, size_t ws_size,
                              hipStream_t stream) {
    const float* x      = (const float*)d_in[0];
    const float* w      = (const float*)d_in[1];
    // d_in[2] = scale (unused in forward)
    const float* in_max = (const float*)d_in[3];
    const float* in_min = (const float*)d_in[4];
    float* out = (float*)d_out;

    const int nRows  = in_sizes[0] / 16;          // B
    const int nTiles = nRows / 16;                // 16-row tiles
    const int waves  = (nTiles + TPW - 1) / TPW;
    const int blocks = (waves + WAVES_PER_BLOCK - 1) / WAVES_PER_BLOCK;

    quantlinear_wmma_kernel<<<blocks, WAVES_PER_BLOCK * 32, 0, stream>>>(
        x, w, in_max, in_min, out, nTiles);
}